// EquivSetGNN_49658411876807
// MI455X (gfx1250) — compile-verified
//
#include <hip/hip_runtime.h>
#include <hip/hip_bf16.h>

// ---------------------------------------------------------------------------
// EquivSetGNN forward for MI455X (gfx1250, wave32, WMMA).
// GEMMs: V_WMMA_F32_16X16X4_F32 (full fp32, matches reference precision).
// Concat-GEMM split: concat([x[v], xe[e]]) @ w2
//   == (x @ w2_top)[v] + (xe @ w2_bot + b2)[e]
// Vertex segment-sum split: xv[v] = deg(v)*t1[v] + scatter(t2[edges])
// ---------------------------------------------------------------------------

#define HDIM   256
#define FIN    512
#define NVERT  50000
#define ESEG   80000
#define NNZ_   400000
#define COUT   40
#define ALPHA_ 0.5f

typedef __attribute__((ext_vector_type(2))) float v2f;
typedef __attribute__((ext_vector_type(8))) float v8f;

#define GEMM_THREADS 512          // 16 waves: block = 16 rows x 256 cols

// C = act( (c1*A1 + c2*A2) @ W + bias ).  A:[M,K], W:[K,N] row-major.
// Requires M % 16 == 0 (true for 50000 / 80000). One block owns its 16 rows
// exclusively (its 16 waves cover all N for N<=256), so Cmat==A1 in-place is safe.
// No load masking: garbage in clamped B columns only affects unstored C cols.
template<int K, int N>
__global__ __launch_bounds__(GEMM_THREADS) void gemm_f32_wmma(
    const float* __restrict__ A1, const float* __restrict__ A2,
    float c1, float c2,
    const float* __restrict__ W, const float* __restrict__ bias,
    float* __restrict__ Cmat, int M, int do_relu)
{
    constexpr int LDSS = K + 2;            // conflict-free ds_load_b64 stride
    __shared__ float aTile[16 * LDSS];
    const int row0 = blockIdx.x * 16;

    // ---- cooperative stage of A strip [16 x K] (restart fusion optional) ----
    if (A2) {
        for (int idx = threadIdx.x; idx < 16 * K; idx += GEMM_THREADS) {
            int r = idx / K, c = idx - r * K;
            size_t g = (size_t)(row0 + r) * K + c;
            aTile[r * LDSS + c] = c1 * A1[g] + c2 * A2[g];
        }
    } else {
        for (int idx = threadIdx.x; idx < 16 * K; idx += GEMM_THREADS) {
            int r = idx / K, c = idx - r * K;
            aTile[r * LDSS + c] = A1[(size_t)(row0 + r) * K + c];
        }
    }
    __syncthreads();

    const int wave = threadIdx.x >> 5;
    const int lane = threadIdx.x & 31;
    const int col0 = wave * 16;
    if (col0 >= N) return;                 // wave-uniform exit (after barrier)

    const int half = lane >> 4;            // 0 -> K=0,1 ; 1 -> K=2,3
    const int ln   = lane & 15;            // A: row-in-tile, B/C: col-in-tile
    const int col  = col0 + ln;
    const int colC = (col < N) ? col : (N - 1);   // address clamp only

    const float* __restrict__ aRow = &aTile[ln * LDSS + 2 * half];
    const float* __restrict__ wp   = W + (size_t)(2 * half) * N + colC;

    v8f acc = {};
    #pragma unroll 8
    for (int k = 0; k < K; k += 4) {
        // A 16x4 fp32: lane(half,ln) holds A[ln][2h], A[ln][2h+1]
        v2f a = *(const v2f*)(aRow + k);
        // B 4x16 fp32: lane(half,ln) holds B[2h][ln], B[2h+1][ln]
        v2f b;
        b.x = wp[0];
        b.y = wp[N];
        wp += 4 * N;
        acc = __builtin_amdgcn_wmma_f32_16x16x4_f32(
                  false, a, false, b, (short)0, acc, false, false);
    }

    if (col < N) {
        float bv = bias ? bias[col] : 0.0f;
        float* __restrict__ cp = Cmat + (size_t)(row0 + 8 * half) * N + col;
        #pragma unroll
        for (int r = 0; r < 8; ++r) {      // C VGPR r -> rows r / r+8
            float v = acc[r] + bv;
            if (do_relu) v = v > 0.0f ? v : 0.0f;
            cp[(size_t)r * N] = v;
        }
    }
}

__global__ __launch_bounds__(256) void zero_f32(float* __restrict__ p, size_t n)
{
    size_t i = (size_t)blockIdx.x * 256 + threadIdx.x;
    if (i < n) p[i] = 0.0f;
}

__global__ __launch_bounds__(256) void zero_i32(int* __restrict__ p, int n)
{
    int i = blockIdx.x * 256 + threadIdx.x;
    if (i < n) p[i] = 0;
}

// deg[vertex[i]] += 1
__global__ __launch_bounds__(256) void count_deg(
    const int* __restrict__ vertex, int* __restrict__ deg, int nnz)
{
    int i = blockIdx.x * 256 + threadIdx.x;
    if (i < nnz) atomicAdd(&deg[vertex[i]], 1);
}

// xe[edges[i], :] += y1[vertex[i], :]   (64 threads per incidence, float4)
__global__ __launch_bounds__(256) void scatter_edge_sum(
    const float* __restrict__ y1, const int* __restrict__ vertex,
    const int* __restrict__ edges, float* __restrict__ xe, int nnz)
{
    size_t tid = (size_t)blockIdx.x * 256 + threadIdx.x;
    int i  = (int)(tid >> 6);
    int c4 = (int)(tid & 63) * 4;
    if (i >= nnz) return;
    int v = vertex[i];
    int e = edges[i];
    float4 val = *(const float4*)&y1[(size_t)v * HDIM + c4];
    float* dst = &xe[(size_t)e * HDIM + c4];
    atomicAdd(dst + 0, val.x);
    atomicAdd(dst + 1, val.y);
    atomicAdd(dst + 2, val.z);
    atomicAdd(dst + 3, val.w);
}

// xv[r, :] = deg[r] * t1[r, :]          (dense init; replaces zero + gather)
__global__ __launch_bounds__(256) void init_deg_scaled(
    const float* __restrict__ t1, const int* __restrict__ deg,
    float* __restrict__ xv, int n4 /* rows*64 */)
{
    int idx = blockIdx.x * 256 + threadIdx.x;
    if (idx >= n4) return;
    int row = idx >> 6;
    float d = (float)deg[row];
    float4 v = ((const float4*)t1)[idx];
    float4 o; o.x = d * v.x; o.y = d * v.y; o.z = d * v.z; o.w = d * v.w;
    ((float4*)xv)[idx] = o;
}

// xv[vertex[i], :] += t2[edges[i], :]
__global__ __launch_bounds__(256) void scatter_vertex_t2(
    const float* __restrict__ t2, const int* __restrict__ vertex,
    const int* __restrict__ edges, float* __restrict__ xv, int nnz)
{
    size_t tid = (size_t)blockIdx.x * 256 + threadIdx.x;
    int i  = (int)(tid >> 6);
    int c4 = (int)(tid & 63) * 4;
    if (i >= nnz) return;
    int v = vertex[i];
    int e = edges[i];
    float4 b = *(const float4*)&t2[(size_t)e * HDIM + c4];
    float* dst = &xv[(size_t)v * HDIM + c4];
    atomicAdd(dst + 0, b.x);
    atomicAdd(dst + 1, b.y);
    atomicAdd(dst + 2, b.z);
    atomicAdd(dst + 3, b.w);
}

extern "C" void kernel_launch(void* const* d_in, const int* in_sizes, int n_in,
                              void* d_out, int out_size, void* d_ws, size_t ws_size,
                              hipStream_t stream)
{
    (void)in_sizes; (void)n_in; (void)out_size; (void)ws_size;

    const float* x_in     = (const float*)d_in[0];   // [N, 512]
    const int*   vertex   = (const int*)  d_in[1];   // [NNZ]
    const int*   edges    = (const int*)  d_in[2];   // [NNZ]
    const float* lin_in_w = (const float*)d_in[3];   // [512, 256]
    const float* lin_in_b = (const float*)d_in[4];
    const float* w1       = (const float*)d_in[5];   // [256, 256]
    const float* b1       = (const float*)d_in[6];
    const float* w2       = (const float*)d_in[7];   // [512, 256]
    const float* b2       = (const float*)d_in[8];
    const float* w3       = (const float*)d_in[9];   // [256, 256]
    const float* b3       = (const float*)d_in[10];
    const float* cls_w1   = (const float*)d_in[11];  // [256, 256]
    const float* cls_b1   = (const float*)d_in[12];
    const float* cls_w2   = (const float*)d_in[13];  // [256, 40]
    const float* cls_b2   = (const float*)d_in[14];
    float* out = (float*)d_out;                       // [N, 40]

    const size_t NH = (size_t)NVERT * HDIM;
    const size_t EH = (size_t)ESEG  * HDIM;
    float* B0  = (float*)d_ws;     // x0
    float* P1  = B0 + NH;          // xv / xnext (layer 0)
    float* P2  = P1 + NH;          // xv / xnext (layer 1)
    float* Y   = P2 + NH;          // y1 / t1 / classifier hidden
    float* XE  = Y  + NH;          // xe / t2 (in-place GEMM)
    int*   DEG = (int*)(XE + EH);  // vertex degrees

    const float* w2_top = w2;
    const float* w2_bot = w2 + (size_t)HDIM * HDIM;

    dim3 gN((NVERT + 15) / 16), gE((ESEG + 15) / 16);
    const unsigned nnzBlocks = (unsigned)(((size_t)NNZ_ * 64 + 255) / 256);

    // degrees (reused by both layers)
    zero_i32<<<(NVERT + 255) / 256, 256, 0, stream>>>(DEG, NVERT);
    count_deg<<<(NNZ_ + 255) / 256, 256, 0, stream>>>(vertex, DEG, NNZ_);

    // 1) x0 = relu(x @ lin_in_w + lin_in_b)
    gemm_f32_wmma<FIN, HDIM><<<gN, GEMM_THREADS, 0, stream>>>(
        x_in, nullptr, 1.0f, 0.0f, lin_in_w, lin_in_b, B0, NVERT, 1);

    const float* xc = B0;
    float* xv_bufs[2] = {P1, P2};

    for (int l = 0; l < 2; ++l) {
        float* XV = xv_bufs[l];

        // 2) y1 = x @ w1 + b1
        gemm_f32_wmma<HDIM, HDIM><<<gN, GEMM_THREADS, 0, stream>>>(
            xc, nullptr, 1.0f, 0.0f, w1, b1, Y, NVERT, 0);

        // 3) xe = segment_sum(y1[vertex], edges)
        zero_f32<<<(unsigned)((EH + 255) / 256), 256, 0, stream>>>(XE, EH);
        scatter_edge_sum<<<nnzBlocks, 256, 0, stream>>>(Y, vertex, edges, XE, NNZ_);

        // 4) t1 = x @ w2_top
        gemm_f32_wmma<HDIM, HDIM><<<gN, GEMM_THREADS, 0, stream>>>(
            xc, nullptr, 1.0f, 0.0f, w2_top, nullptr, Y, NVERT, 0);

        // 5) t2 = xe @ w2_bot + b2   (in-place on XE)
        gemm_f32_wmma<HDIM, HDIM><<<gE, GEMM_THREADS, 0, stream>>>(
            XE, nullptr, 1.0f, 0.0f, w2_bot, b2, XE, ESEG, 0);

        // 6) xv = deg * t1 + segment_sum(t2[edges], vertex)
        init_deg_scaled<<<(NVERT * 64 + 255) / 256, 256, 0, stream>>>(
            Y, DEG, XV, NVERT * 64);
        scatter_vertex_t2<<<nnzBlocks, 256, 0, stream>>>(XE, vertex, edges, XV, NNZ_);

        // 7) x = relu( ((1-a)*xv + a*x0) @ w3 + b3 )   (in-place on XV)
        gemm_f32_wmma<HDIM, HDIM><<<gN, GEMM_THREADS, 0, stream>>>(
            XV, B0, 1.0f - ALPHA_, ALPHA_, w3, b3, XV, NVERT, 1);

        xc = XV;
    }

    // 8) h = relu(x @ cls_w1 + cls_b1)
    gemm_f32_wmma<HDIM, HDIM><<<gN, GEMM_THREADS, 0, stream>>>(
        xc, nullptr, 1.0f, 0.0f, cls_w1, cls_b1, Y, NVERT, 1);

    // 9) out = h @ cls_w2 + cls_b2   (N=40, clamped cols, guarded store)
    gemm_f32_wmma<HDIM, COUT><<<gN, GEMM_THREADS, 0, stream>>>(
        Y, nullptr, 1.0f, 0.0f, cls_w2, cls_b2, out, NVERT, 0);
}